// OurButterflyTopKSAE_82660940579335
// MI455X (gfx1250) — compile-verified
//
#include <hip/hip_runtime.h>
#include <hip/hip_bf16.h>

// ---------------------------------------------------------------------------
// Fused Butterfly-TopK-SAE for gfx1250 (MI455X).
//  - 1 workgroup = 4 batch rows, full N=16384 state resident in LDS (256KB).
//  - Butterfly stages 0-11 applied as per-block dense 16x16 matrices via
//    V_WMMA_F32_16X16X4_F32 (fp32, K=16 as 4 chained K=4 WMMAs).
//    A rows 4..15 carry duplicated data (their D rows are discarded), so all
//    operand loads are unconditional; B matrices are packed per-lane so each
//    block needs only 2x global_load_b128.
//  - Stages 12-13 elementwise in LDS.
//  - Top-64 per row via 4-pass 8-bit radix select on IEEE bits in LDS.
//  - enc_rows permutation folded via LDS u16 inverse table; decode input is
//    a sparse scatter of <=64 values per row.
// ---------------------------------------------------------------------------

#define ACT        1024
#define NN         16384     // N == DICT
#define LOGN       14
#define BROWS      4         // batch rows per workgroup
#define TPB        1024      // 32 waves
#define TOPK_K     64
#define LIST_CAP   96

typedef __attribute__((ext_vector_type(2))) float v2f;
typedef __attribute__((ext_vector_type(4))) float v4f;
typedef __attribute__((ext_vector_type(8))) float v8f;

// ---- dynamic shared memory layout (bytes) ----
#define SM_V      0                              // 4*16384 f32      = 262144
#define SM_INV    (SM_V    + BROWS*NN*4)         // 16384 u16        =  32768
#define SM_HIST   (SM_INV  + NN*2)               // 4*256 u32        =   4096
#define SM_S1     (SM_HIST + BROWS*256*4)        // 1024 f32         =   4096
#define SM_S2     (SM_S1   + TPB*4)              // 1024 f32         =   4096
#define SM_MEAN   (SM_S2   + TPB*4)              // 4 f32
#define SM_STD    (SM_MEAN + 16)
#define SM_PREF   (SM_STD  + 16)                 // 4 u32
#define SM_KK     (SM_PREF + 16)                 // 4 i32
#define SM_CNT    (SM_KK   + 16)                 // 4 u32
#define SM_LVAL   (SM_CNT  + 16)                 // 4*96 f32 = 1536
#define SM_LPOS   (SM_LVAL + BROWS*LIST_CAP*4)   // 4*96 u16 = 768
#define SM_TOTAL  (SM_LPOS + BROWS*LIST_CAP*2)   // = 309584 <= 320KB

// ---------------------------------------------------------------------------
// Precompute: collapse butterfly stage-group g (stages 4g..4g+3) into dense
// 16x16 matrices, one per block b (1024 blocks). Stored in per-lane-packed
// WMMA-B order: lane L = half*16 + n owns 8 consecutive floats, holding the
// B elements (k = c*4 + 2*half + e, n) at [L*8 + c*2 + e].
// ---------------------------------------------------------------------------
__global__ void sae_build_group_mats(const float* __restrict__ params,
                                     float* __restrict__ W) {
    const int b = blockIdx.x;          // 0..1023
    const int g = blockIdx.y;          // 0..2
    const int u = threadIdx.x & 15;    // input index  (B k-index)
    const int t = threadIdx.x >> 4;    // output index (B n-index)
    __shared__ float M[16][16];

    const int S    = 1 << (4 * g);
    const int base = ((b >> (4 * g)) << (4 * (g + 1))) | (b & (S - 1));

    float cur = (t == u) ? 1.0f : 0.0f;
    M[t][u] = cur;
    __syncthreads();

#pragma unroll
    for (int l = 0; l < 4; ++l) {
        const int i  = 4 * g + l;           // global stage index
        const int tx = t ^ (1 << l);
        float other  = M[tx][u];
        float a  = params[(2 * i)     * NN + base + S * t];
        float bb = params[(2 * i + 1) * NN + base + S * tx];
        __syncthreads();
        cur = a * cur + bb * other;         // v'[t] = a[j(t)]v[t] + b[j(t^e)]v[t^e]
        M[t][u] = cur;
        __syncthreads();
    }
    // per-lane-packed B layout: k = u = c*4 + 2*half + e
    const int c    = u >> 2;
    const int half = (u >> 1) & 1;
    const int e    = u & 1;
    const int L    = half * 16 + t;
    W[((g << 10) + b) * 256 + L * 8 + c * 2 + e] = cur;
}

// ---------------------------------------------------------------------------
// Apply stage groups 0..2 (stages 0..11) to LDS-resident v via WMMA.
// Each wave owns disjoint blocks; blocks partition all N positions per group,
// so only one barrier per group is needed.
// ---------------------------------------------------------------------------
__device__ __forceinline__ void sae_apply_groups(float* __restrict__ v,
                                                 const float* __restrict__ W) {
    const int tid  = threadIdx.x;
    const int wave = tid >> 5;       // 0..31
    const int lane = tid & 31;
    const int m    = lane & 15;      // M index (padded batch row)
    const int half = lane >> 4;      // 0/1
    const int vrow = (m & 3) * NN;   // rows 4..15 duplicate rows 0..3 (D rows discarded)

    for (int g = 0; g < 3; ++g) {
        const int S = 1 << (4 * g);
        for (int b = wave; b < 1024; b += 32) {
            const int base = ((b >> (4 * g)) << (4 * (g + 1))) | (b & (S - 1));
            const float* Wb = W + ((g << 10) + b) * 256 + lane * 8;
            if (b + 32 < 1024)
                __builtin_prefetch(Wb + 32 * 256, 0, 0);

            // B operand: 8 consecutive floats per lane -> 2x global_load_b128
            const v4f bq0 = *(const v4f*)(Wb);
            const v4f bq1 = *(const v4f*)(Wb + 4);

            // A operand: 8 LDS reads, all unconditional
            float aa[8];
#pragma unroll
            for (int c = 0; c < 4; ++c) {
                const int k0 = c * 4 + half * 2;
                aa[2 * c]     = v[vrow + base + S * k0];
                aa[2 * c + 1] = v[vrow + base + S * (k0 + 1)];
            }

            v8f acc = {};
            v2f A, B;
            A.x = aa[0]; A.y = aa[1]; B.x = bq0[0]; B.y = bq0[1];
            acc = __builtin_amdgcn_wmma_f32_16x16x4_f32(false, A, false, B, (short)0, acc, false, false);
            A.x = aa[2]; A.y = aa[3]; B.x = bq0[2]; B.y = bq0[3];
            acc = __builtin_amdgcn_wmma_f32_16x16x4_f32(false, A, false, B, (short)0, acc, false, false);
            A.x = aa[4]; A.y = aa[5]; B.x = bq1[0]; B.y = bq1[1];
            acc = __builtin_amdgcn_wmma_f32_16x16x4_f32(false, A, false, B, (short)0, acc, false, false);
            A.x = aa[6]; A.y = aa[7]; B.x = bq1[2]; B.y = bq1[3];
            acc = __builtin_amdgcn_wmma_f32_16x16x4_f32(false, A, false, B, (short)0, acc, false, false);

            // D: VGPR r, lanes 0-15 -> (M=r, N=lane). Rows 0..3 are real.
            if (half == 0) {
#pragma unroll
                for (int r = 0; r < BROWS; ++r)
                    v[r * NN + base + S * m] = acc[r];
            }
        }
        __syncthreads();
    }
}

// ---------------------------------------------------------------------------
// Stages 12,13 (strides 4096, 8192): elementwise butterfly in LDS.
// One thread owns both halves of each pair -> in-place safe.
// ---------------------------------------------------------------------------
__device__ __forceinline__ void sae_apply_hi_stages(float* __restrict__ v,
                                                    const float* __restrict__ p) {
    const int tid = threadIdx.x;
#pragma unroll
    for (int st = 0; st < 2; ++st) {
        const int bit = 12 + st;
        const int s   = 1 << bit;
        const float* pa = p + (2 * (12 + st))     * NN;
        const float* pb = p + (2 * (12 + st) + 1) * NN;
        for (int i = tid; i < BROWS * (NN / 2); i += TPB) {
            const int r  = i / (NN / 2);
            const int q  = i % (NN / 2);
            const int j  = ((q >> bit) << (bit + 1)) | (q & (s - 1));
            const int j2 = j | s;
            const float va = v[r * NN + j];
            const float vb = v[r * NN + j2];
            const float o1 = pa[j]  * va + pb[j2] * vb;
            const float o2 = pa[j2] * vb + pb[j]  * va;
            v[r * NN + j]  = o1;
            v[r * NN + j2] = o2;
        }
        __syncthreads();
    }
}

// ---------------------------------------------------------------------------
// Main fused kernel: 1024 blocks x 1024 threads, 4 batch rows per block.
// ---------------------------------------------------------------------------
__global__ void __launch_bounds__(TPB, 1)
sae_fused(const float* __restrict__ x, const float* __restrict__ b_dec,
          const float* __restrict__ enc_p, const float* __restrict__ dec_p,
          const int* __restrict__ enc_rows, const int* __restrict__ dec_rows,
          const float* __restrict__ Wenc, const float* __restrict__ Wdec,
          float* __restrict__ out) {
    extern __shared__ unsigned char smem[];
    float*          v     = (float*)(smem + SM_V);
    unsigned short* invp  = (unsigned short*)(smem + SM_INV);
    unsigned*       hist  = (unsigned*)(smem + SM_HIST);
    float*          s1    = (float*)(smem + SM_S1);
    float*          s2    = (float*)(smem + SM_S2);
    float*          meanv = (float*)(smem + SM_MEAN);
    float*          stdv  = (float*)(smem + SM_STD);
    unsigned*       pref  = (unsigned*)(smem + SM_PREF);
    int*            kk    = (int*)(smem + SM_KK);
    unsigned*       cnt   = (unsigned*)(smem + SM_CNT);
    float*          lval  = (float*)(smem + SM_LVAL);
    unsigned short* lpos  = (unsigned short*)(smem + SM_LPOS);

    const int tid  = threadIdx.x;
    const int row0 = blockIdx.x * BROWS;

    // ---- 1. per-row mean / std (ddof=1) ----
    {
        const int r = tid >> 8;        // 0..3 (256 threads per row)
        const int t = tid & 255;
        float s = 0.0f, q = 0.0f;
        const float* xr = x + (size_t)(row0 + r) * ACT;
        for (int i = t; i < ACT; i += 256) { float val = xr[i]; s += val; q += val * val; }
        s1[tid] = s; s2[tid] = q;
        __syncthreads();
        for (int off = 128; off > 0; off >>= 1) {
            if (t < off) { s1[tid] += s1[tid + off]; s2[tid] += s2[tid + off]; }
            __syncthreads();
        }
        if (t == 0) {
            const float sum = s1[tid], sq = s2[tid];
            const float mean = sum / (float)ACT;
            float var = (sq - sum * mean) / (float)(ACT - 1);
            var = fmaxf(var, 0.0f);
            meanv[r] = mean;
            stdv[r]  = sqrtf(var);
        }
        __syncthreads();
    }

    // ---- 2. init v = [(x-mean)/(std+1e-5) - b_dec, 0...] ; build inverse perm ----
    for (int i = tid; i < BROWS * NN; i += TPB) {
        const int r = i >> 14, j = i & (NN - 1);
        float val = 0.0f;
        if (j < ACT)
            val = (x[(size_t)(row0 + r) * ACT + j] - meanv[r]) / (stdv[r] + 1e-5f) - b_dec[j];
        v[i] = val;
    }
    for (int m = tid; m < NN; m += TPB)
        invp[enc_rows[m]] = (unsigned short)m;
    __syncthreads();

    // ---- 3. encode butterfly ----
    sae_apply_groups(v, Wenc);          // stages 0..11 (WMMA)
    sae_apply_hi_stages(v, enc_p);      // stages 12,13

    // relu (in-place)
    for (int i = tid; i < BROWS * NN; i += TPB)
        v[i] = fmaxf(v[i], 0.0f);
    __syncthreads();

    // ---- 4. exact 64th-largest per row via 4x8-bit radix select on IEEE bits ----
    if (tid < BROWS) { pref[tid] = 0u; kk[tid] = TOPK_K; }
    __syncthreads();
    for (int bp = 3; bp >= 0; --bp) {
        for (int i = tid; i < BROWS * 256; i += TPB) hist[i] = 0u;
        __syncthreads();
        const unsigned known = (bp == 3) ? 0u : (0xFFFFFFFFu << ((bp + 1) * 8));
        for (int i = tid; i < BROWS * NN; i += TPB) {
            const int r = i >> 14;
            const unsigned u = __float_as_uint(v[i]);
            if (u != 0u && (u & known) == pref[r])
                atomicAdd(&hist[r * 256 + ((u >> (bp * 8)) & 255u)], 1u);
        }
        __syncthreads();
        if (tid < BROWS) {
            const int r = tid;
            int need = kk[r];
            unsigned acc = 0; int sel = 0;
            for (int bin = 255; bin >= 0; --bin) {
                const unsigned c = hist[r * 256 + bin];
                if (acc + c >= (unsigned)need) { sel = bin; need -= (int)acc; break; }
                acc += c;
            }
            kk[r]   = need;
            pref[r] |= ((unsigned)sel) << (bp * 8);
        }
        __syncthreads();
    }

    // ---- 5. compact survivors (value, permuted position) ----
    if (tid < BROWS) cnt[tid] = 0u;
    __syncthreads();
    for (int i = tid; i < BROWS * NN; i += TPB) {
        const int r = i >> 14, j = i & (NN - 1);
        const float val = v[i];
        const unsigned u = __float_as_uint(val);
        if (u != 0u && u >= pref[r]) {
            const unsigned p = atomicAdd(&cnt[r], 1u);
            if (p < LIST_CAP) {
                lpos[r * LIST_CAP + p] = invp[j];   // decode-input position
                lval[r * LIST_CAP + p] = val;
            }
        }
    }
    __syncthreads();

    // ---- 6. rebuild v as sparse decode input ----
    for (int i = tid; i < BROWS * NN; i += TPB) v[i] = 0.0f;
    __syncthreads();
    for (int i = tid; i < BROWS * LIST_CAP; i += TPB) {
        const int r = i / LIST_CAP, p = i % LIST_CAP;
        unsigned c = cnt[r]; if (c > LIST_CAP) c = LIST_CAP;
        if ((unsigned)p < c)
            v[r * NN + lpos[r * LIST_CAP + p]] = lval[r * LIST_CAP + p];
    }
    __syncthreads();

    // ---- 7. decode butterfly ----
    sae_apply_groups(v, Wdec);
    sae_apply_hi_stages(v, dec_p);

    // ---- 8. gather dec_rows, denormalize, write out ----
    for (int i = tid; i < BROWS * ACT; i += TPB) {
        const int r = i >> 10, c = i & (ACT - 1);
        const float rec = v[r * NN + dec_rows[c]] + b_dec[c];
        out[(size_t)(row0 + r) * ACT + c] = rec * stdv[r] + meanv[r];
    }
}

// ---------------------------------------------------------------------------
extern "C" void kernel_launch(void* const* d_in, const int* in_sizes, int n_in,
                              void* d_out, int out_size, void* d_ws, size_t ws_size,
                              hipStream_t stream) {
    (void)in_sizes; (void)n_in; (void)out_size; (void)ws_size;
    const float* x        = (const float*)d_in[0];
    const float* b_dec    = (const float*)d_in[1];
    const float* enc_p    = (const float*)d_in[2];
    const float* dec_p    = (const float*)d_in[3];
    const int*   enc_rows = (const int*)d_in[4];
    const int*   dec_rows = (const int*)d_in[5];
    float*       out      = (float*)d_out;

    float* Wenc = (float*)d_ws;                  // 3*1024*256 f32 = 3 MB
    float* Wdec = Wenc + 3 * 1024 * 256;         // 3 MB

    dim3 gw(1024, 3, 1);
    sae_build_group_mats<<<gw, 256, 0, stream>>>(enc_p, Wenc);
    sae_build_group_mats<<<gw, 256, 0, stream>>>(dec_p, Wdec);

    sae_fused<<<dim3(4096 / BROWS), dim3(TPB), SM_TOTAL, stream>>>(
        x, b_dec, enc_p, dec_p, enc_rows, dec_rows, Wenc, Wdec, out);
}